// RPNLoss_19988777795705
// MI455X (gfx1250) — compile-verified
//
#include <hip/hip_runtime.h>
#include <hip/hip_bf16.h>

#define N_IMG   2
#define M_GT    64
#define K_ANCH  159882
#define LOW_T   0.3f
#define HIGH_T  0.7f
#define SL1BETA (1.0f / 9.0f)

struct Ptrs {
    const float* cls[5];
    const float* reg[5];
};

// level decode: anchor index k (within image) -> level, plane offset (h*W+w), anchor a, HW
__device__ __forceinline__ void decode_level(int k, int& lvl, int& plane, int& a, int& HW) {
    int off;
    if (k < 120000)      { lvl = 0; off = 0;      HW = 40000; }
    else if (k < 150000) { lvl = 1; off = 120000; HW = 10000; }
    else if (k < 157500) { lvl = 2; off = 150000; HW = 2500;  }
    else if (k < 159375) { lvl = 3; off = 157500; HW = 625;   }
    else                 { lvl = 4; off = 159375; HW = 169;   }
    int rem = k - off;
    a     = rem % 3;      // fastest-varying is anchor (permute order H,W,A)
    plane = rem / 3;      // = h*W + w
}

__device__ __forceinline__ float4 load_anchor_box(const Ptrs& p, int img, int k) {
    int lvl, plane, a, HW;
    decode_level(k, lvl, plane, a, HW);
    const float* r = p.reg[lvl];
    size_t base = (size_t)(img * 12 + a * 4) * (size_t)HW + (size_t)plane;
    float4 b;
    b.x = r[base];
    b.y = r[base + (size_t)HW];
    b.z = r[base + 2 * (size_t)HW];
    b.w = r[base + 3 * (size_t)HW];
    return b;
}

__device__ __forceinline__ float load_cls_logit(const Ptrs& p, int img, int k) {
    int lvl, plane, a, HW;
    decode_level(k, lvl, plane, a, HW);
    return p.cls[lvl][(size_t)(img * 3 + a) * (size_t)HW + (size_t)plane];
}

// noinline: single codegen instance -> bit-identical IoU in pass1 and pass2,
// so the force-match float equality behaves exactly like the JAX reference.
__device__ __noinline__ float iou_one(float4 ab, float4 g) {
    float area1 = (g.z - g.x) * (g.w - g.y);      // gt area
    float area2 = (ab.z - ab.x) * (ab.w - ab.y);  // anchor (reg pred) area
    float ltx = fmaxf(g.x, ab.x);
    float lty = fmaxf(g.y, ab.y);
    float rbx = fminf(g.z, ab.z);
    float rby = fminf(g.w, ab.w);
    float w = fmaxf(rbx - ltx, 0.0f);
    float h = fmaxf(rby - lty, 0.0f);
    float inter = w * h;
    return inter / (area1 + area2 - inter);
}

// CDNA5 async global -> LDS copy (ASYNCcnt-tracked DMA path)
__device__ __forceinline__ void async_gt_to_lds(float* s_gt, const float* gt_img, int tid) {
    if (tid < M_GT) {
        unsigned lds_off = (unsigned)(size_t)(&s_gt[tid * 4]);
        const float* src = gt_img + tid * 4;
        asm volatile("global_load_async_to_lds_b128 %0, %1, off"
                     :: "v"(lds_off), "v"(src) : "memory");
    }
    asm volatile("s_wait_asynccnt 0" ::: "memory");
}

__global__ void rpn_init(double* sums, unsigned* best) {
    int t = threadIdx.x;
    if (t < 2) sums[t] = 0.0;
    if (t < N_IMG * M_GT) best[t] = 0u;  // 0.0f bits; all IoUs are >= 0
}

__global__ __launch_bounds__(256) void rpn_pass1(Ptrs p, const float* __restrict__ gt,
                                                 unsigned* __restrict__ best) {
    __shared__ __align__(16) float s_gt[M_GT * 4];
    __shared__ unsigned s_best[M_GT];
    const int tid = threadIdx.x;
    const int img = blockIdx.y;

    async_gt_to_lds(s_gt, gt + (size_t)img * M_GT * 4, tid);
    if (tid < M_GT) s_best[tid] = 0u;
    __syncthreads();

    int k = blockIdx.x * 256 + tid;
    bool valid = (k < K_ANCH);
    float4 ab = load_anchor_box(p, img, valid ? k : (K_ANCH - 1));

    for (int m = 0; m < M_GT; ++m) {
        float4 g = *(const float4*)(&s_gt[m * 4]);
        float v = iou_one(ab, g);
        if (!valid) v = 0.0f;
        // wave32 max reduction
        for (int off = 16; off > 0; off >>= 1)
            v = fmaxf(v, __shfl_xor(v, off, 32));
        if ((tid & 31) == 0)
            atomicMax(&s_best[m], __float_as_uint(v));  // uint order == float order (vals >= 0)
    }
    __syncthreads();
    if (tid < M_GT)
        atomicMax(&best[img * M_GT + tid], s_best[tid]);
}

__global__ __launch_bounds__(256) void rpn_pass2(Ptrs p, const float* __restrict__ gt,
                                                 const unsigned* __restrict__ best,
                                                 double* __restrict__ sums) {
    __shared__ __align__(16) float s_gt[M_GT * 4];
    __shared__ float s_bpg[M_GT];
    const int tid = threadIdx.x;
    const int img = blockIdx.y;

    async_gt_to_lds(s_gt, gt + (size_t)img * M_GT * 4, tid);
    if (tid < M_GT) s_bpg[tid] = __uint_as_float(best[img * M_GT + tid]);
    __syncthreads();

    int k = blockIdx.x * 256 + tid;
    bool valid = (k < K_ANCH);
    int kk = valid ? k : (K_ANCH - 1);
    float4 ab = load_anchor_box(p, img, kk);
    float x = load_cls_logit(p, img, kk);

    float bestv = -INFINITY;
    int bestm = 0;
    bool force = false;
    for (int m = 0; m < M_GT; ++m) {
        float4 g = *(const float4*)(&s_gt[m * 4]);
        float v = iou_one(ab, g);
        if (v > bestv) { bestv = v; bestm = m; }   // strict > == jnp.argmax first-max
        force = force || (v == s_bpg[m]);          // exact equality, bit-identical recompute
    }

    int match;
    if (force)               match = bestm;
    else if (bestv < LOW_T)  match = -1;
    else if (bestv < HIGH_T) match = -2;
    else                     match = bestm;

    float label = (match == -2) ? -1.0f : ((match >= 0) ? 1.0f : 0.0f);
    int mi = (match >= 0) ? match : 0;             // clip(matches, 0)

    // BCE with logits (per-element)
    float clsT = fmaxf(x, 0.0f) - x * label + log1pf(expf(-fabsf(x)));

    // smooth L1 over 4 coords vs matched gt
    float pr[4] = { ab.x, ab.y, ab.z, ab.w };
    const float* gm = &s_gt[mi * 4];
    float regT = 0.0f;
#pragma unroll
    for (int j = 0; j < 4; ++j) {
        float d = fabsf(pr[j] - gm[j]);
        regT += (d < SL1BETA) ? (0.5f * d * d) / SL1BETA : (d - 0.5f * SL1BETA);
    }
    if (!valid) { clsT = 0.0f; regT = 0.0f; }

    // wave32 sum reductions
    for (int off = 16; off > 0; off >>= 1) {
        clsT += __shfl_xor(clsT, off, 32);
        regT += __shfl_xor(regT, off, 32);
    }
    if ((tid & 31) == 0) {
        atomicAdd(&sums[0], (double)clsT);
        atomicAdd(&sums[1], (double)regT);
    }
}

__global__ void rpn_fin(const double* sums, float* out) {
    double n_cls = (double)N_IMG * (double)K_ANCH;       // 319764
    double n_reg = n_cls * 4.0;                          // 1279056
    out[0] = (float)(sums[0] / n_cls + sums[1] / n_reg);
}

extern "C" void kernel_launch(void* const* d_in, const int* in_sizes, int n_in,
                              void* d_out, int out_size, void* d_ws, size_t ws_size,
                              hipStream_t stream) {
    // setup_inputs() dict order is interleaved: cls_l0, reg_l0, cls_l1, reg_l1, ..., gt_boxes
    Ptrs p;
    for (int i = 0; i < 5; ++i) {
        p.cls[i] = (const float*)d_in[2 * i];
        p.reg[i] = (const float*)d_in[2 * i + 1];
    }
    const float* gt = (const float*)d_in[10];

    double*   sums = (double*)d_ws;                       // [0]=cls_sum, [1]=reg_sum
    unsigned* best = (unsigned*)((char*)d_ws + 16);       // N_IMG*M_GT float bits

    rpn_init<<<1, 128, 0, stream>>>(sums, best);

    dim3 grid((K_ANCH + 255) / 256, N_IMG);
    rpn_pass1<<<grid, 256, 0, stream>>>(p, gt, best);
    rpn_pass2<<<grid, 256, 0, stream>>>(p, gt, best, sums);

    rpn_fin<<<1, 1, 0, stream>>>(sums, (float*)d_out);
}